// GCN_gen_64630667870457
// MI455X (gfx1250) — compile-verified
//
#include <hip/hip_runtime.h>
#include <hip/hip_bf16.h>

typedef float v2f __attribute__((ext_vector_type(2)));
typedef float v8f __attribute__((ext_vector_type(8)));

// ---------------------------------------------------------------------------
// elementwise helpers
// ---------------------------------------------------------------------------
__global__ void zero_f32(float* __restrict__ p, int n4) {
  int i = blockIdx.x * blockDim.x + threadIdx.x;
  if (i < n4) ((float4*)p)[i] = make_float4(0.f, 0.f, 0.f, 0.f);
}

__global__ void degree_kernel(const int* __restrict__ src, const int* __restrict__ dst,
                              float* __restrict__ degOut, float* __restrict__ degIn, int nE) {
  int e = blockIdx.x * blockDim.x + threadIdx.x;
  if (e < nE) {
    atomicAdd(&degOut[src[e]], 1.0f);
    atomicAdd(&degIn[dst[e]], 1.0f);
  }
}

__global__ void norm_kernel(float* __restrict__ deg, int n) {
  int i = blockIdx.x * blockDim.x + threadIdx.x;
  if (i < n) {
    float d = fmaxf(deg[i], 1.0f);
    deg[i] = 1.0f / sqrtf(d);   // clip(deg,1)^-0.5
  }
}

// ---------------------------------------------------------------------------
// Pack W[K,N] into WMMA B-fragment order: pair p = (g*2+hi)*N + col holds
// { W[4g+2hi][col], W[4g+2hi+1][col] } so the GEMM B load is one b64.
// ---------------------------------------------------------------------------
__global__ void pack_w_kernel(const float* __restrict__ W, float* __restrict__ Wp,
                              int K, int N, int total) {
  int gid = blockIdx.x * blockDim.x + threadIdx.x;   // over (K/4)*2*N pairs
  if (gid >= total) return;
  int col  = gid % N;
  int rest = gid / N;
  int hi = rest & 1;
  int g  = rest >> 1;
  int k  = 4 * g + 2 * hi;
  Wp[2 * gid + 0] = W[(size_t)k * N + col];
  Wp[2 * gid + 1] = W[(size_t)(k + 1) * N + col];
}

// ---------------------------------------------------------------------------
// GEMM: Hp[M,N] = (A[M,K] * norm_s[M]) @ W[K,N]  via V_WMMA_F32_16X16X4_F32.
// K, N compile-time -> all fragment addresses are base + immediate.
// K-loop processes chunks of 4 K-steps: 20 independent b64 loads issued
// ahead of 16 WMMAs so the backend can clause loads with staggered waits.
// Block = 4 waves sharing one 16-col stripe (blockIdx.y); each wave owns a
// 64x16 tile. norm_s applied in the epilogue (rows of D are independent).
// ---------------------------------------------------------------------------
template <int K, int N>
__global__ void __launch_bounds__(128)
gemm_norm_wmma(const float* __restrict__ A, const float* __restrict__ Wp,
               const float* __restrict__ ns, float* __restrict__ Hp, int M) {
  const int wave   = threadIdx.x >> 5;
  const int lane   = threadIdx.x & 31;
  const int laneLo = lane & 15;
  const int laneHi = lane >> 4;               // 0 | 1

  const int row0 = blockIdx.x * 256 + wave * 64;  // wave-uniform
  if (row0 >= M) return;                          // wave-uniform exit
  const int col = blockIdx.y * 16 + laneLo;

  // Clamp OOB rows (garbage rows are masked at store; rows are independent).
  const float* ap0, *ap1, *ap2, *ap3;
  float nsr[4];
  {
    const float* aps[4];
    #pragma unroll
    for (int t = 0; t < 4; ++t) {
      int rr = row0 + t * 16 + laneLo;
      int rc = rr < M ? rr : (M - 1);
      nsr[t] = rr < M ? ns[rr] : 0.0f;
      aps[t] = A + (size_t)rc * K;
    }
    ap0 = aps[0]; ap1 = aps[1]; ap2 = aps[2]; ap3 = aps[3];
  }

  const v2f* wp = (const v2f*)Wp + (size_t)laneHi * N + col;  // + g*2N per step
  const int kOff = laneHi << 1;
  constexpr int K4 = K >> 2;                 // number of 16x16x4 K-steps

  v8f acc0 = {}, acc1 = {}, acc2 = {}, acc3 = {};
  #pragma unroll 1
  for (int gg = 0; gg < K4; gg += 4) {
    // ---- issue all 20 loads for this chunk (distinct SSA values) ----
    v2f b[4];
    v2f a0[4], a1[4], a2[4], a3[4];
    #pragma unroll
    for (int u = 0; u < 4; ++u) b[u] = wp[2 * N * (gg + u)];
    const int ka = (gg << 2) + kOff;
    #pragma unroll
    for (int u = 0; u < 4; ++u) {
      a0[u] = *(const v2f*)(ap0 + ka + 4 * u);
      a1[u] = *(const v2f*)(ap1 + ka + 4 * u);
      a2[u] = *(const v2f*)(ap2 + ka + 4 * u);
      a3[u] = *(const v2f*)(ap3 + ka + 4 * u);
    }
    // ---- 16 WMMAs consuming the batch ----
    #pragma unroll
    for (int u = 0; u < 4; ++u) {
      acc0 = __builtin_amdgcn_wmma_f32_16x16x4_f32(false, a0[u], false, b[u], (short)0, acc0, false, false);
      acc1 = __builtin_amdgcn_wmma_f32_16x16x4_f32(false, a1[u], false, b[u], (short)0, acc1, false, false);
      acc2 = __builtin_amdgcn_wmma_f32_16x16x4_f32(false, a2[u], false, b[u], (short)0, acc2, false, false);
      acc3 = __builtin_amdgcn_wmma_f32_16x16x4_f32(false, a3[u], false, b[u], (short)0, acc3, false, false);
    }
  }

  // C/D layout: VGPR v, lanes 0-15 -> M=v, lanes 16-31 -> M=v+8.
  // Row scale ns[row] recovered by shuffling nsr (lane r holds row r's ns).
  v8f accs[4] = {acc0, acc1, acc2, acc3};
  #pragma unroll
  for (int t = 0; t < 4; ++t) {
    int rowBase = row0 + t * 16 + 8 * laneHi;
    #pragma unroll
    for (int v = 0; v < 8; ++v) {
      int row = rowBase + v;
      float sc = __shfl(nsr[t], v + 8 * laneHi, 32);
      if (row < M) Hp[(size_t)row * N + col] = accs[t][v] * sc;
    }
  }
}

// ---------------------------------------------------------------------------
// Scatter: Agg[dst[e]] += Hp[src[e]]  (b128 gather + f32 atomics, L2-resident)
// ---------------------------------------------------------------------------
__global__ void scatter_add_kernel(const float* __restrict__ Hp, const int* __restrict__ src,
                                   const int* __restrict__ dst, float* __restrict__ Agg,
                                   int nE, int N, int shift) {
  int gid = blockIdx.x * blockDim.x + threadIdx.x;
  int e = gid >> shift;
  if (e >= nE) return;
  int f = (gid & ((1 << shift) - 1)) << 2;
  int s = src[e];
  int d = dst[e];
  float4 v = *(const float4*)(Hp + (size_t)s * N + f);
  float* out = Agg + (size_t)d * N + f;
  atomicAdd(out + 0, v.x);
  atomicAdd(out + 1, v.y);
  atomicAdd(out + 2, v.z);
  atomicAdd(out + 3, v.w);
}

// ---------------------------------------------------------------------------
// Finalize: Out = Agg * norm_d[row] + b  (+ ReLU)
// ---------------------------------------------------------------------------
__global__ void finalize_kernel(const float* __restrict__ Agg, const float* __restrict__ nd,
                                const float* __restrict__ bias, float* __restrict__ Out,
                                int total4, int shift, int doRelu) {
  int gid = blockIdx.x * blockDim.x + threadIdx.x;
  if (gid >= total4) return;
  int row = gid >> shift;
  int c   = (gid & ((1 << shift) - 1)) << 2;
  float s = nd[row];
  float4 a = ((const float4*)Agg)[gid];
  float4 b = *(const float4*)(bias + c);
  float4 r;
  r.x = a.x * s + b.x;  r.y = a.y * s + b.y;
  r.z = a.z * s + b.z;  r.w = a.w * s + b.w;
  if (doRelu) {
    r.x = fmaxf(r.x, 0.f); r.y = fmaxf(r.y, 0.f);
    r.z = fmaxf(r.z, 0.f); r.w = fmaxf(r.w, 0.f);
  }
  ((float4*)Out)[gid] = r;
}

// ---------------------------------------------------------------------------
// Host orchestration
// ---------------------------------------------------------------------------
extern "C" void kernel_launch(void* const* d_in, const int* in_sizes, int n_in,
                              void* d_out, int out_size, void* d_ws, size_t ws_size,
                              hipStream_t stream) {
  const float* features = (const float*)d_in[0];
  const int*   src      = (const int*)d_in[1];
  const int*   dst      = (const int*)d_in[2];
  const float* Ws[4] = {(const float*)d_in[3], (const float*)d_in[5],
                        (const float*)d_in[7], (const float*)d_in[9]};
  const float* Bs[4] = {(const float*)d_in[4], (const float*)d_in[6],
                        (const float*)d_in[8], (const float*)d_in[10]};

  const int dims[5] = {512, 256, 256, 128, 64};
  const int M    = in_sizes[0] / dims[0];
  const int nE   = in_sizes[1];
  const int Mpad = ((M + 63) / 64) * 64;

  float* ws     = (float*)d_ws;
  float* norm_s = ws;                                  // Mpad
  float* norm_d = norm_s + Mpad;                       // Mpad
  float* hp     = norm_d + Mpad;                       // Mpad*256
  float* bufA   = hp   + (size_t)Mpad * 256;           // Mpad*256
  float* bufB   = bufA + (size_t)Mpad * 256;           // Mpad*256
  float* wpk    = bufB + (size_t)Mpad * 256;           // packed weights
  float* Wp[4];
  {
    float* p = wpk;
    for (int i = 0; i < 4; ++i) { Wp[i] = p; p += (size_t)dims[i] * dims[i + 1]; }
  }

  float* outp   = (float*)d_out;
  float* aspect = outp;                                // M x 128
  float* logits = outp + (size_t)M * dims[3];          // M x 64

  // degrees -> norms (both arrays contiguous)
  int z4 = (2 * Mpad) / 4;
  zero_f32<<<(z4 + 255) / 256, 256, 0, stream>>>(norm_s, z4);
  degree_kernel<<<(nE + 255) / 256, 256, 0, stream>>>(src, dst, norm_s, norm_d, nE);
  norm_kernel<<<(2 * Mpad + 255) / 256, 256, 0, stream>>>(norm_s, 2 * Mpad);

  // pack all weights into WMMA B-fragment order
  for (int i = 0; i < 4; ++i) {
    int K = dims[i], N = dims[i + 1];
    int pairs = (K / 4) * 2 * N;
    pack_w_kernel<<<(pairs + 255) / 256, 256, 0, stream>>>(Ws[i], Wp[i], K, N, pairs);
  }

  auto ilog2 = [](int x) { int s = 0; while ((1 << s) < x) ++s; return s; };

  auto post_gemm = [&](int N, int li, float* agg, float* outb, bool relu) {
    int a4 = (M * N) / 4;
    zero_f32<<<(a4 + 255) / 256, 256, 0, stream>>>(agg, a4);
    int shiftE = ilog2(N / 4);
    long long total = (long long)nE << shiftE;
    int blocks = (int)((total + 255) / 256);
    scatter_add_kernel<<<blocks, 256, 0, stream>>>(hp, src, dst, agg, nE, N, shiftE);
    finalize_kernel<<<(a4 + 255) / 256, 256, 0, stream>>>(
        agg, norm_d, Bs[li], outb, a4, shiftE, relu ? 1 : 0);
  };

  const int gmx = (M + 255) / 256;

  // L0: in(512) -> hidden(256), relu
  gemm_norm_wmma<512, 256><<<dim3(gmx, 256 / 16), 128, 0, stream>>>(features, Wp[0], norm_s, hp, M);
  post_gemm(256, 0, bufA, bufA, true);

  // L1: hidden(256) -> hidden(256), relu
  gemm_norm_wmma<256, 256><<<dim3(gmx, 256 / 16), 128, 0, stream>>>(bufA, Wp[1], norm_s, hp, M);
  post_gemm(256, 1, bufB, bufB, true);

  // L2: hidden(256) -> aspect(128), no act
  gemm_norm_wmma<256, 128><<<dim3(gmx, 128 / 16), 128, 0, stream>>>(bufB, Wp[2], norm_s, hp, M);
  post_gemm(128, 2, bufA, aspect, false);

  // L3: aspect(128) -> classes(64), no act
  gemm_norm_wmma<128, 64><<<dim3(gmx, 64 / 16), 128, 0, stream>>>(aspect, Wp[3], norm_s, hp, M);
  post_gemm(64, 3, bufA, logits, false);
}